// RationalInterpolation2D_17282948399684
// MI455X (gfx1250) — compile-verified
//
#include <hip/hip_runtime.h>

// RationalInterpolation2D on MI455X (gfx1250):
//   stage1: g = (values*w2 @ inv2^T) / (w2 @ inv2^T)      -> dual-accumulator WMMA GEMM
//   stage2: out = ((inv1*w1) @ g) / den1[:,None]           -> WMMA GEMM + row scale
// fp32 throughout (barycentric 1/(x-node) factors need full precision), so the
// matrix path is V_WMMA_F32_16X16X4_F32.

#define N1c 512
#define N2c 512
#define M1c 8192
#define M2c 8192
#define EPS_F 1e-14f
#define PI_F 3.14159265358979323846f

typedef float v2f __attribute__((ext_vector_type(2)));
typedef float v8f __attribute__((ext_vector_type(8)));

__device__ __forceinline__ v8f wmma4(v2f a, v2f b, v8f c) {
  // (neg_a, A, neg_b, B, c_mod, C, reuse_a, reuse_b)
  return __builtin_amdgcn_wmma_f32_16x16x4_f32(false, a, false, b, (short)0, c,
                                               false, false);
}

__device__ __forceinline__ float cheb_node(int n, float inv_nm1) {
  return cosf(PI_F * (float)n * inv_nm1);
}

// ---------------------------------------------------------------- prep kernels

// hit[m] = any_n |x[m] - node[n]| < eps   (Chebyshev-Lobatto nodes cos(pi*n/(NN-1)))
__global__ void hits_kernel(const float* __restrict__ x, int M, int NN,
                            int* __restrict__ hit) {
  int m = blockIdx.x * blockDim.x + threadIdx.x;
  if (m >= M) return;
  float xm = x[m];
  float s = 1.0f / (float)(NN - 1);
  int h = 0;
  for (int n = 0; n < NN; ++n) {
    float node = cheb_node(n, s);
    h |= (fabsf(xm - node) < EPS_F) ? 1 : 0;
  }
  hit[m] = h;
}

// inv2T[n, m] = hit2[m] ? (|d|<eps ? 1 : 0) : 1/d,  d = x2[m] - node2[n]
__global__ void build_inv2T_kernel(const float* __restrict__ x2,
                                   const int* __restrict__ hit2,
                                   float* __restrict__ inv2T) {
  int m = blockIdx.x * blockDim.x + threadIdx.x;   // 0..M2-1 (coalesced)
  int n = blockIdx.y;                              // 0..N2-1
  float node = cheb_node(n, 1.0f / (float)(N2c - 1));
  float d = x2[m] - node;
  float inv = hit2[m] ? ((fabsf(d) < EPS_F) ? 1.0f : 0.0f) : (1.0f / d);
  inv2T[(size_t)n * M2c + m] = inv;
}

__device__ __forceinline__ float cheb_w1(int n) {
  float w = (n & 1) ? -1.0f : 1.0f;
  if (n == 0 || n == N1c - 1) w *= 0.5f;
  return w;
}

// den1[m] = sum_n inv1[m,n] * w1[n]
__global__ void den1_kernel(const float* __restrict__ x1,
                            const int* __restrict__ hit1,
                            float* __restrict__ den1) {
  int m = blockIdx.x * blockDim.x + threadIdx.x;
  if (m >= M1c) return;
  float xm = x1[m];
  int h = hit1[m];
  float den = 0.0f;
  for (int n = 0; n < N1c; ++n) {
    float node = cheb_node(n, 1.0f / (float)(N1c - 1));
    float d = xm - node;
    float inv = h ? ((fabsf(d) < EPS_F) ? 1.0f : 0.0f) : (1.0f / d);
    den += inv * cheb_w1(n);
  }
  den1[m] = den;
}

// a1[m, n] = inv1[m,n] * w1[n]   (row-major M1 x N1, coalesced over n)
__global__ void build_a1_kernel(const float* __restrict__ x1,
                                const int* __restrict__ hit1,
                                float* __restrict__ a1) {
  int n = blockIdx.x * blockDim.x + threadIdx.x;   // 0..N1-1
  int m = blockIdx.y;                              // 0..M1-1
  float node = cheb_node(n, 1.0f / (float)(N1c - 1));
  float d = x1[m] - node;
  float inv = hit1[m] ? ((fabsf(d) < EPS_F) ? 1.0f : 0.0f) : (1.0f / d);
  a1[(size_t)m * N1c + n] = inv * cheb_w1(n);
}

// ---------------------------------------------------------------- WMMA GEMMs

#define TBM 64
#define TBN 128
#define TBK 32
#define LDS_A 36  // pad: keeps float4 (16B) staging alignment + spreads LDS banks

// GEMM1: g[k,m] = (vw @ B)/(w @ B), A tiles built on the fly from values*weights2
// and weights2 (dual accumulators share the B fragments).
__global__ __launch_bounds__(256) void gemm1_interp_kernel(
    const float* __restrict__ values, const float* __restrict__ weights2,
    const float* __restrict__ B /* inv2T: N2 x M2 */,
    float* __restrict__ g /* N1 x M2 */) {
  __shared__ float Avw[TBM * LDS_A];
  __shared__ float Aw[TBM * LDS_A];
  __shared__ float Bs[TBK * TBN];

  const int tid = threadIdx.x;
  const int bn = blockIdx.x * TBN;
  const int bm = blockIdx.y * TBM;
  const int lane = tid & 31, wv = tid >> 5;
  const int wrow = wv >> 2, wcol = wv & 3;     // 2 x 4 waves of 32x32
  const int lrow = lane & 15, kh = lane >> 4;  // half-wave K split

  v8f accN[2][2] = {};
  v8f accD[2][2] = {};

  for (int k0 = 0; k0 < N2c; k0 += TBK) {
    // stage A tiles (TBM x TBK): 2 float4 per thread per matrix
    for (int i = tid; i < TBM * (TBK / 4); i += 256) {
      int r = i / (TBK / 4);
      int c = (i - r * (TBK / 4)) * 4;
      const float4 vv = *(const float4*)&values[(size_t)(bm + r) * N2c + k0 + c];
      const float4 ww = *(const float4*)&weights2[(size_t)(bm + r) * N2c + k0 + c];
      float4 p; p.x = vv.x * ww.x; p.y = vv.y * ww.y; p.z = vv.z * ww.z; p.w = vv.w * ww.w;
      *(float4*)&Avw[r * LDS_A + c] = p;
      *(float4*)&Aw[r * LDS_A + c] = ww;
    }
    // stage B tile (TBK x TBN): 4 float4 per thread
    for (int i = tid; i < TBK * (TBN / 4); i += 256) {
      int r = i / (TBN / 4);
      int c = (i - r * (TBN / 4)) * 4;
      *(float4*)&Bs[r * TBN + c] =
          *(const float4*)&B[(size_t)(k0 + r) * M2c + bn + c];
    }
    if (k0 + TBK < N2c)  // global_prefetch_b8 of next B chunk
      __builtin_prefetch(&B[(size_t)(k0 + TBK + (tid >> 5)) * M2c + bn + lane * 4], 0, 1);
    __syncthreads();

#pragma unroll
    for (int kk = 0; kk < TBK; kk += 4) {
      const int ka = kk + 2 * kh;  // ISA 16x4 A layout: lanes 16-31 hold K+2
      v2f av0 = *(const v2f*)&Avw[(wrow * 32 + lrow) * LDS_A + ka];
      v2f av1 = *(const v2f*)&Avw[(wrow * 32 + 16 + lrow) * LDS_A + ka];
      v2f aw0 = *(const v2f*)&Aw[(wrow * 32 + lrow) * LDS_A + ka];
      v2f aw1 = *(const v2f*)&Aw[(wrow * 32 + 16 + lrow) * LDS_A + ka];
      const int bc = wcol * 32 + lrow;
      v2f b0, b1;
      b0.x = Bs[ka * TBN + bc];      b0.y = Bs[(ka + 1) * TBN + bc];
      b1.x = Bs[ka * TBN + bc + 16]; b1.y = Bs[(ka + 1) * TBN + bc + 16];
      accN[0][0] = wmma4(av0, b0, accN[0][0]);
      accN[0][1] = wmma4(av0, b1, accN[0][1]);
      accN[1][0] = wmma4(av1, b0, accN[1][0]);
      accN[1][1] = wmma4(av1, b1, accN[1][1]);
      accD[0][0] = wmma4(aw0, b0, accD[0][0]);
      accD[0][1] = wmma4(aw0, b1, accD[0][1]);
      accD[1][0] = wmma4(aw1, b0, accD[1][0]);
      accD[1][1] = wmma4(aw1, b1, accD[1][1]);
    }
    __syncthreads();
  }

#pragma unroll
  for (int i = 0; i < 2; ++i) {
    const int rbase = bm + wrow * 32 + i * 16 + 8 * kh;  // C/D: VGPR v holds M=v / v+8
#pragma unroll
    for (int j = 0; j < 2; ++j) {
      const int col = bn + wcol * 32 + j * 16 + lrow;
#pragma unroll
      for (int v = 0; v < 8; ++v)
        g[(size_t)(rbase + v) * M2c + col] = accN[i][j][v] / accD[i][j][v];
    }
  }
}

// GEMM2: out[m1, m2] = (a1 @ g)[m1, m2] / den1[m1]
__global__ __launch_bounds__(256) void gemm2_scaled_kernel(
    const float* __restrict__ A /* a1: M1 x N1 */,
    const float* __restrict__ B /* g:  N1 x M2 */,
    const float* __restrict__ den1, float* __restrict__ out) {
  __shared__ float As[TBM * LDS_A];
  __shared__ float Bs[TBK * TBN];

  const int tid = threadIdx.x;
  const int bn = blockIdx.x * TBN;
  const int bm = blockIdx.y * TBM;
  const int lane = tid & 31, wv = tid >> 5;
  const int wrow = wv >> 2, wcol = wv & 3;
  const int lrow = lane & 15, kh = lane >> 4;

  v8f acc[2][2] = {};

  for (int k0 = 0; k0 < N1c; k0 += TBK) {
    for (int i = tid; i < TBM * (TBK / 4); i += 256) {
      int r = i / (TBK / 4);
      int c = (i - r * (TBK / 4)) * 4;
      *(float4*)&As[r * LDS_A + c] =
          *(const float4*)&A[(size_t)(bm + r) * N1c + k0 + c];
    }
    for (int i = tid; i < TBK * (TBN / 4); i += 256) {
      int r = i / (TBN / 4);
      int c = (i - r * (TBN / 4)) * 4;
      *(float4*)&Bs[r * TBN + c] =
          *(const float4*)&B[(size_t)(k0 + r) * M2c + bn + c];
    }
    if (k0 + TBK < N1c)
      __builtin_prefetch(&B[(size_t)(k0 + TBK + (tid >> 5)) * M2c + bn + lane * 4], 0, 1);
    __syncthreads();

#pragma unroll
    for (int kk = 0; kk < TBK; kk += 4) {
      const int ka = kk + 2 * kh;
      v2f a0 = *(const v2f*)&As[(wrow * 32 + lrow) * LDS_A + ka];
      v2f a1f = *(const v2f*)&As[(wrow * 32 + 16 + lrow) * LDS_A + ka];
      const int bc = wcol * 32 + lrow;
      v2f b0, b1;
      b0.x = Bs[ka * TBN + bc];      b0.y = Bs[(ka + 1) * TBN + bc];
      b1.x = Bs[ka * TBN + bc + 16]; b1.y = Bs[(ka + 1) * TBN + bc + 16];
      acc[0][0] = wmma4(a0, b0, acc[0][0]);
      acc[0][1] = wmma4(a0, b1, acc[0][1]);
      acc[1][0] = wmma4(a1f, b0, acc[1][0]);
      acc[1][1] = wmma4(a1f, b1, acc[1][1]);
    }
    __syncthreads();
  }

#pragma unroll
  for (int i = 0; i < 2; ++i) {
    const int rbase = bm + wrow * 32 + i * 16 + 8 * kh;
    float dv[8];
#pragma unroll
    for (int v = 0; v < 8; ++v) dv[v] = den1[rbase + v];
#pragma unroll
    for (int j = 0; j < 2; ++j) {
      const int col = bn + wcol * 32 + j * 16 + lrow;
#pragma unroll
      for (int v = 0; v < 8; ++v)
        out[(size_t)(rbase + v) * M2c + col] = acc[i][j][v] / dv[v];
    }
  }
}

// ---------------------------------------------------------------- launch

extern "C" void kernel_launch(void* const* d_in, const int* in_sizes, int n_in,
                              void* d_out, int out_size, void* d_ws, size_t ws_size,
                              hipStream_t stream) {
  const float* x1 = (const float*)d_in[0];        // (M1,)
  const float* x2 = (const float*)d_in[1];        // (M2,)
  const float* values = (const float*)d_in[2];    // (N1, N2)
  const float* weights2 = (const float*)d_in[3];  // (N1, N2)
  float* out = (float*)d_out;                     // (M1, M2)

  // workspace layout (~48.1 MB)
  float* ws = (float*)d_ws;
  float* inv2T = ws;                              // N2 * M2
  float* g = inv2T + (size_t)N2c * M2c;           // N1 * M2
  float* a1 = g + (size_t)N1c * M2c;              // M1 * N1
  float* den1 = a1 + (size_t)M1c * N1c;           // M1
  int* hit2 = (int*)(den1 + M1c);                 // M2
  int* hit1 = hit2 + M2c;                         // M1

  hits_kernel<<<M2c / 256, 256, 0, stream>>>(x2, M2c, N2c, hit2);
  hits_kernel<<<M1c / 256, 256, 0, stream>>>(x1, M1c, N1c, hit1);
  build_inv2T_kernel<<<dim3(M2c / 256, N2c), 256, 0, stream>>>(x2, hit2, inv2T);
  den1_kernel<<<M1c / 256, 256, 0, stream>>>(x1, hit1, den1);
  build_a1_kernel<<<dim3(N1c / 256, M1c), 256, 0, stream>>>(x1, hit1, a1);

  gemm1_interp_kernel<<<dim3(M2c / TBN, N1c / TBM), 256, 0, stream>>>(
      values, weights2, inv2T, g);
  gemm2_scaled_kernel<<<dim3(M2c / TBN, M1c / TBM), 256, 0, stream>>>(
      a1, g, den1, out);
}